// KGATLayer_67259187855785
// MI455X (gfx1250) — compile-verified
//
#include <hip/hip_runtime.h>
#include <math.h>

#define EMBED 128

typedef __attribute__((ext_vector_type(2))) float v2f;
typedef __attribute__((ext_vector_type(8))) float v8f;

// ---- native f32 global atomic add (global_atomic_add_f32, no CAS loop) ----
__device__ __forceinline__ void atomic_add_f32(float* p, float v) {
    unsafeAtomicAdd(p, v);
}

// ---- monotone uint encoding of float for atomicMax ----
__device__ __forceinline__ unsigned int encode_max(float f) {
    unsigned int u = __float_as_uint(f);
    return (u & 0x80000000u) ? ~u : (u | 0x80000000u);
}
__device__ __forceinline__ float decode_max(unsigned int k) {
    return (k & 0x80000000u) ? __uint_as_float(k ^ 0x80000000u)
                             : __uint_as_float(~k);
}

// ============================================================
// K1: per-edge score = sum(e_t * tanh(e_h + e_r)); global max.
// One wave (32 lanes) per edge; each lane owns 4 of the 128 dims.
// ============================================================
__global__ void kgat_edge_score(const float* __restrict__ ent,
                                const float* __restrict__ rel,
                                const int* __restrict__ heads,
                                const int* __restrict__ rels,
                                const int* __restrict__ tails,
                                float* __restrict__ score,
                                unsigned int* __restrict__ gmax_enc,
                                int n_edges) {
    __shared__ float wscore[8];
    const int wave = threadIdx.x >> 5;
    const int lane = threadIdx.x & 31;
    const int edge = blockIdx.x * 8 + wave;

    float s = -3.402823466e+38f;
    if (edge < n_edges) {
        const int h = heads[edge];
        const int r = rels[edge];
        const int t = tails[edge];
        const float4 eh = *(const float4*)(ent + (size_t)h * EMBED + lane * 4);
        const float4 et = *(const float4*)(ent + (size_t)t * EMBED + lane * 4);
        const float4 er = *(const float4*)(rel + (size_t)r * EMBED + lane * 4);
        float p = et.x * tanhf(eh.x + er.x)
                + et.y * tanhf(eh.y + er.y)
                + et.z * tanhf(eh.z + er.z)
                + et.w * tanhf(eh.w + er.w);
#pragma unroll
        for (int off = 16; off > 0; off >>= 1)
            p += __shfl_xor(p, off, 32);
        if (lane == 0) score[edge] = p;
        s = p;
    }
    if (lane == 0) wscore[wave] = s;
    __syncthreads();
    if (threadIdx.x == 0) {
        float m = wscore[0];
#pragma unroll
        for (int i = 1; i < 8; ++i) m = fmaxf(m, wscore[i]);
        atomicMax(gmax_enc, encode_max(m));
    }
}

// ============================================================
// K2: score_exp = exp(score - gmax); segment-sum into score_sum[head].
// ============================================================
__global__ void kgat_edge_exp(float* __restrict__ score,
                              const int* __restrict__ heads,
                              const unsigned int* __restrict__ gmax_enc,
                              float* __restrict__ score_sum,
                              int n_edges) {
    const int e = blockIdx.x * blockDim.x + threadIdx.x;
    if (e >= n_edges) return;
    const float gmax = decode_max(*gmax_enc);
    const float se = expf(score[e] - gmax);
    score[e] = se;  // overwrite with numerator
    atomic_add_f32(&score_sum[heads[e]], se);
}

// ============================================================
// K3: agg[head] += attn * e_t  (wave per edge, float4 per lane,
// 4 native f32 global atomics per lane).  agg lives in d_out.
// ============================================================
__global__ void kgat_aggregate(const float* __restrict__ ent,
                               const int* __restrict__ heads,
                               const int* __restrict__ tails,
                               const float* __restrict__ score_exp,
                               const float* __restrict__ score_sum,
                               float* __restrict__ agg,
                               int n_edges) {
    const int wave = threadIdx.x >> 5;
    const int lane = threadIdx.x & 31;
    const int edge = blockIdx.x * 8 + wave;
    if (edge >= n_edges) return;

    const int h = heads[edge];
    const int t = tails[edge];
    const float attn = score_exp[edge] / (score_sum[h] + 1e-10f);
    const float4 et = *(const float4*)(ent + (size_t)t * EMBED + lane * 4);
    float* dst = agg + (size_t)h * EMBED + lane * 4;
    atomic_add_f32(dst + 0, attn * et.x);
    atomic_add_f32(dst + 1, attn * et.y);
    atomic_add_f32(dst + 2, attn * et.z);
    atomic_add_f32(dst + 3, attn * et.w);
}

// ============================================================
// K4: out = LeakyReLU( (entity_emb + agg) @ W^T ), in place on d_out.
// Block = 256 threads = 8 waves; block owns 16 rows; wave w owns
// output columns [16w, 16w+16).  K-loop over 128 in steps of 4 with
// V_WMMA_F32_16X16X4_F32.
//
// f32 16x16x4 A layout (MxK): lane l holds M=l&15,
//   VGPR0/1 = K = k0 + 2*(l>>4) + {0,1}           -> float2 from LDS
// B layout (KxN), B[k][n] = W[n0+n][k]: lane l holds N=l&15,
//   VGPR0/1 = K = k0 + 2*(l>>4) + {0,1}           -> float2 from W row
// C/D layout: vgpr j -> M = j + 8*(l>>4), N = l&15.
// ============================================================
__global__ void kgat_gemm_lrelu(const float* __restrict__ ent,
                                const float* __restrict__ W,
                                float* __restrict__ out,
                                int n_rows) {
    __shared__ float Xs[16][EMBED + 4];  // +4 floats: 16B-aligned rows, bank spread

    const int block_row = blockIdx.x * 16;
    const int tid = threadIdx.x;

    // Stage X = entity_emb + agg for 16 rows (2048 floats, float4 granular)
    for (int i = tid; i < 16 * (EMBED / 4); i += 256) {
        const int r = i >> 5;          // 32 float4 per row
        const int c = (i & 31) * 4;
        const int row = block_row + r;
        float4 x = make_float4(0.f, 0.f, 0.f, 0.f);
        if (row < n_rows) {
            const float4 e = *(const float4*)(ent + (size_t)row * EMBED + c);
            const float4 a = *(const float4*)(out + (size_t)row * EMBED + c);
            x = make_float4(e.x + a.x, e.y + a.y, e.z + a.z, e.w + a.w);
        }
        *(float4*)(&Xs[r][c]) = x;
    }
    __syncthreads();

    const int wave = tid >> 5;
    const int lane = tid & 31;
    const int n0 = wave * 16;           // output column tile
    const int mi = lane & 15;           // row (A) / column (B) within tile
    const int khalf = (lane >> 4) * 2;  // K sub-offset per half-wave

    v8f c = {};
#pragma unroll
    for (int k0 = 0; k0 < EMBED; k0 += 4) {
        v2f a;
        a.x = Xs[mi][k0 + khalf];
        a.y = Xs[mi][k0 + khalf + 1];
        const float* wp = W + (size_t)(n0 + mi) * EMBED + k0 + khalf;
        v2f b;
        b.x = wp[0];
        b.y = wp[1];
        c = __builtin_amdgcn_wmma_f32_16x16x4_f32(
                /*neg_a=*/false, a, /*neg_b=*/false, b,
                /*c_mod=*/(short)0, c, /*reuse_a=*/false, /*reuse_b=*/false);
    }

    // Write back with LeakyReLU(0.2)
    const int mbase = (lane >> 4) * 8;
#pragma unroll
    for (int j = 0; j < 8; ++j) {
        const int row = block_row + mbase + j;
        if (row < n_rows) {
            float v = c[j];
            v = (v >= 0.f) ? v : 0.2f * v;
            out[(size_t)row * EMBED + n0 + mi] = v;
        }
    }
}

// ============================================================
// Launch pipeline
// ============================================================
extern "C" void kernel_launch(void* const* d_in, const int* in_sizes, int n_in,
                              void* d_out, int out_size, void* d_ws, size_t ws_size,
                              hipStream_t stream) {
    const float* ent  = (const float*)d_in[0];
    const float* rel  = (const float*)d_in[1];
    const float* W    = (const float*)d_in[2];
    const int* heads  = (const int*)d_in[3];
    const int* rels   = (const int*)d_in[4];
    const int* tails  = (const int*)d_in[5];

    const int n_edges    = in_sizes[3];
    const int n_entities = in_sizes[0] / EMBED;
    float* out = (float*)d_out;

    // workspace layout: [score/score_exp: E][score_sum: N][gmax_enc: 1]
    float* score            = (float*)d_ws;
    float* score_sum        = score + n_edges;
    unsigned int* gmax_enc  = (unsigned int*)(score_sum + n_entities);

    // zero score_sum + gmax slot (encoded 0 == smaller than any float key),
    // and zero d_out which serves as the agg accumulator
    hipMemsetAsync(score_sum, 0, (size_t)(n_entities + 1) * sizeof(float), stream);
    hipMemsetAsync(out, 0, (size_t)out_size * sizeof(float), stream);

    const int edge_wave_blocks = (n_edges + 7) / 8;      // 8 waves/block, wave/edge
    kgat_edge_score<<<edge_wave_blocks, 256, 0, stream>>>(
        ent, rel, heads, rels, tails, score, gmax_enc, n_edges);

    kgat_edge_exp<<<(n_edges + 255) / 256, 256, 0, stream>>>(
        score, heads, gmax_enc, score_sum, n_edges);

    kgat_aggregate<<<edge_wave_blocks, 256, 0, stream>>>(
        ent, heads, tails, score, score_sum, out, n_edges);

    kgat_gemm_lrelu<<<(n_entities + 15) / 16, 256, 0, stream>>>(
        ent, W, out, n_entities);
}